// HNetEmbeddingsSTFT_28312424415323
// MI455X (gfx1250) — compile-verified
//
#include <hip/hip_runtime.h>
#include <hip/hip_bf16.h>
#include <math.h>

typedef __attribute__((ext_vector_type(2))) float v2f;
typedef __attribute__((ext_vector_type(8))) float v8f;

#define V      256
#define NFREQ  129
#define TFR    17        // frames
#define LSEQ   1024
#define KC     33024     // NFREQ * V
#define MROWS  34        // B * TFR
#define MPAD   48
#define NJ     308
#define NPAD   320
#define DEMB   716

// ---------------------------------------------------------------- kernel 1
// zero the pad rows (34..47) of the dense A matrix
__global__ void zero_pad_kernel(float* feat) {
    size_t i = (size_t)blockIdx.x * 256 + threadIdx.x;
    const size_t total = (size_t)(MPAD - MROWS) * KC;
    if (i < total) feat[(size_t)MROWS * KC + i] = 0.0f;
}

// ---------------------------------------------------------------- kernel 2
// Sparse-delta STFT magnitude -> dense feat[row=b*17+f][k*256+v]
// grid: (34, 17)  block: 256   (blockIdx.y = chunk of 8 freqs)
__global__ __launch_bounds__(256) void stft_feat_kernel(
        const int* __restrict__ ids, const float* __restrict__ window,
        float* __restrict__ feat) {
    __shared__ int   sid[256];
    __shared__ float win[256];
    __shared__ float tab[256];
    __shared__ int   startv[256];
    __shared__ int   cntv[256];
    __shared__ short posl[256];

    const int bf  = blockIdx.x;            // row = b*17 + f
    const int b   = bf / TFR;
    const int f   = bf % TFR;
    const int tid = threadIdx.x;

    // reflect-padded sample -> original index (pad = 128, L = 1024)
    int p = f * 64 + tid;
    int t = (p < 128) ? (128 - p) : ((p < 1152) ? (p - 128) : (2174 - p));
    sid[tid] = ids[b * LSEQ + t];
    win[tid] = window[tid];
    tab[tid] = cosf((float)tid * 0.02454369260617026f); // 2*pi/256
    __syncthreads();

    // deterministic counting sort: bucket positions by id, ascending n in bucket
    const int myid = sid[tid];
    int start_v = 0, cnt_v = 0, start_id = 0, rank = 0;
    for (int n = 0; n < 256; ++n) {
        int s = sid[n];
        start_v  += (s < tid);
        cnt_v    += (s == tid);
        start_id += (s < myid);
        rank     += (s == myid) && (n < tid);
    }
    startv[tid] = start_v;
    cntv[tid]   = cnt_v;
    posl[start_id + rank] = (short)tid;
    __syncthreads();

    const int s0 = startv[tid];
    const int c0 = cntv[tid];
    for (int kk = 0; kk < 8; ++kk) {
        int k = blockIdx.y * 8 + kk;
        if (k >= NFREQ) break;
        float re = 0.0f, im = 0.0f;
        for (int e = 0; e < c0; ++e) {
            int n = posl[s0 + e];
            int j = (k * n) & 255;
            float w = win[n];
            re += w * tab[j];
            im -= w * tab[(j + 192) & 255];   // sin(x) = cos(x - pi/2)
        }
        float mag = sqrtf(re * re + im * im);
        feat[(size_t)bf * KC + k * V + tid] = mag;
    }
}

// ---------------------------------------------------------------- kernel 3
// G[m][n] = sum_c feat[m][c] * proj_w[n][c]   (M=48pad, N=320pad, K=33024)
// grid: (20 Ntiles, 3 Mtiles), block: 256 = 8 waves (wave32); each wave
// covers K/8 = 4128 with v_wmma_f32_16x16x4_f32; deterministic LDS reduce.
__global__ __launch_bounds__(256) void gemm_kernel(
        const float* __restrict__ feat, const float* __restrict__ projw,
        float* __restrict__ G) {
    __shared__ float red[8 * 32 * 8];       // 8 KB

    const int n0   = blockIdx.x * 16;
    const int m0   = blockIdx.y * 16;
    const int lane = threadIdx.x & 31;
    const int wv   = threadIdx.x >> 5;      // 0..7
    const int kh   = lane >> 4;             // K-half select (ISA A/B layout)

    const int m  = m0 + (lane & 15);
    int jn = n0 + (lane & 15);
    if (jn > NJ - 1) jn = NJ - 1;           // clamp; garbage cols never read

    const float* arow = feat  + (size_t)m  * KC + 2 * kh;
    const float* brow = projw + (size_t)jn * KC + 2 * kh;

    const int kbeg = wv * (KC / 8);         // 4128 per wave, %16 == 0
    const int kend = kbeg + (KC / 8);

    v8f acc0 = {0,0,0,0,0,0,0,0}, acc1 = {0,0,0,0,0,0,0,0};
    v8f acc2 = {0,0,0,0,0,0,0,0}, acc3 = {0,0,0,0,0,0,0,0};

    for (int k = kbeg; k < kend; k += 16) {
        v2f a0 = *(const v2f*)(arow + k);
        v2f b0 = *(const v2f*)(brow + k);
        acc0 = __builtin_amdgcn_wmma_f32_16x16x4_f32(false, a0, false, b0,
                                                     (short)0, acc0, false, false);
        v2f a1 = *(const v2f*)(arow + k + 4);
        v2f b1 = *(const v2f*)(brow + k + 4);
        acc1 = __builtin_amdgcn_wmma_f32_16x16x4_f32(false, a1, false, b1,
                                                     (short)0, acc1, false, false);
        v2f a2 = *(const v2f*)(arow + k + 8);
        v2f b2 = *(const v2f*)(brow + k + 8);
        acc2 = __builtin_amdgcn_wmma_f32_16x16x4_f32(false, a2, false, b2,
                                                     (short)0, acc2, false, false);
        v2f a3 = *(const v2f*)(arow + k + 12);
        v2f b3 = *(const v2f*)(brow + k + 12);
        acc3 = __builtin_amdgcn_wmma_f32_16x16x4_f32(false, a3, false, b3,
                                                     (short)0, acc3, false, false);
    }
    v8f acc = (acc0 + acc1) + (acc2 + acc3); // fixed order -> deterministic

#pragma unroll
    for (int r = 0; r < 8; ++r) red[(wv * 32 + lane) * 8 + r] = acc[r];
    __syncthreads();

    if (wv == 0) {
#pragma unroll
        for (int r = 0; r < 8; ++r) {
            float s = red[(0 * 32 + lane) * 8 + r];
            for (int w = 1; w < 8; ++w)       // fixed order -> deterministic
                s += red[(w * 32 + lane) * 8 + r];
            // C/D layout: VGPR r -> M = 8*(lane>=16) + r, N = lane&15
            int mm = m0 + 8 * kh + r;
            int nn = n0 + (lane & 15);
            G[mm * NPAD + nn] = s;
        }
    }
}

// ---------------------------------------------------------------- kernel 4
// out[b][l][0..715] = emb_table[id];  out[b][l][716+j] = lerp(G, t) + bias
__global__ __launch_bounds__(256) void out_kernel(
        const int* __restrict__ ids, const float* __restrict__ emb,
        const float* __restrict__ projb, const float* __restrict__ G,
        float* __restrict__ out) {
    const int bl = blockIdx.x;              // 0..2047
    const int b  = bl >> 10;
    const int l  = bl & 1023;
    const int id = ids[bl];

    float coord = (l + 0.5f) * (17.0f / 1024.0f) - 0.5f;
    coord = fminf(fmaxf(coord, 0.0f), 16.0f);
    int   i0 = (int)floorf(coord);
    int   i1 = (i0 + 1 < TFR) ? i0 + 1 : TFR - 1;
    float w  = coord - (float)i0;

    const float* g0 = G + (b * TFR + i0) * NPAD;
    const float* g1 = G + (b * TFR + i1) * NPAD;
    const float* e  = emb + (size_t)id * DEMB;
    float*       o  = out + (size_t)bl * 1024;

    for (int c = threadIdx.x; c < 1024; c += 256) {
        float val;
        if (c < DEMB) {
            val = e[c];
        } else {
            int j = c - DEMB;
            val = (1.0f - w) * g0[j] + w * g1[j] + projb[j];
        }
        o[c] = val;
    }
}

// ---------------------------------------------------------------- launch
extern "C" void kernel_launch(void* const* d_in, const int* in_sizes, int n_in,
                              void* d_out, int out_size, void* d_ws, size_t ws_size,
                              hipStream_t stream) {
    (void)in_sizes; (void)n_in; (void)out_size; (void)ws_size;
    const int*   ids    = (const int*)  d_in[0];
    const float* emb    = (const float*)d_in[1];
    const float* projw  = (const float*)d_in[2];
    const float* projb  = (const float*)d_in[3];
    const float* window = (const float*)d_in[4];
    float* out  = (float*)d_out;
    float* feat = (float*)d_ws;                       // MPAD x KC  (6.34 MB)
    float* G    = feat + (size_t)MPAD * KC;           // MPAD x NPAD (61 KB)

    zero_pad_kernel<<<((MPAD - MROWS) * KC) / 256, 256, 0, stream>>>(feat);
    stft_feat_kernel<<<dim3(MROWS, 17), 256, 0, stream>>>(ids, window, feat);
    gemm_kernel<<<dim3(NPAD / 16, MPAD / 16), 256, 0, stream>>>(feat, projw, G);
    out_kernel<<<2048, 256, 0, stream>>>(ids, emb, projb, G, out);
}